// TwoLayerModel_78159814852847
// MI455X (gfx1250) — compile-verified
//
#include <hip/hip_runtime.h>
#include <math.h>

typedef __attribute__((ext_vector_type(2))) float v2f;
typedef __attribute__((ext_vector_type(8))) float v8f;

#define EXP_SCALE 0.76f
#define TPB 256

// ---------------------------------------------------------------------------
// deg[i] = 1.0 (self loop)
__global__ void k_fill_deg(float* __restrict__ deg, int n) {
    int i = blockIdx.x * blockDim.x + threadIdx.x;
    if (i < n) deg[i] = 1.0f;
}

// deg[dst[e]] += 1 per edge
__global__ void k_count_deg(const int* __restrict__ dst, float* __restrict__ deg, int E) {
    int e = blockIdx.x * blockDim.x + threadIdx.x;
    if (e < E) atomicAdd(&deg[dst[e]], 1.0f);
}

// deg -> dis = rsqrt(deg) in place (deg >= 1 due to self loop)
__global__ void k_rsqrt(float* __restrict__ deg, int n) {
    int i = blockIdx.x * blockDim.x + threadIdx.x;
    if (i < n) {
        float d = deg[i];
        deg[i] = d > 0.0f ? rsqrtf(d) : 0.0f;
    }
}

// ---------------------------------------------------------------------------
// C[M x 32] = A[M x K] * B[K x 32] using V_WMMA_F32_16X16X4_F32.
// One wave32 per 16x16 output tile. M must be a multiple of 16 (100000 = 6250*16),
// K a multiple of 4. EXEC is all-ones per wave as WMMA requires.
__global__ void __launch_bounds__(32) k_gemm_wmma_f32(const float* __restrict__ A,
                                                      const float* __restrict__ B,
                                                      float* __restrict__ C,
                                                      int K) {
    const int lane = threadIdx.x & 31;
    const int lo = lane & 15;          // M (for A) or N (for B/C) within tile
    const int hi = lane >> 4;          // K-pair selector / C row-half selector
    const int m0 = blockIdx.x << 4;    // row tile base
    const int n0 = blockIdx.y << 4;    // col tile base (N total = 32 -> gridDim.y = 2)

    v8f acc = {};
    const float* arow = A + (size_t)(m0 + lo) * (size_t)K;

    for (int k = 0; k < K; k += 4) {
        const int ka = k + 2 * hi;     // this lane-group's K pair: {ka, ka+1}
        // A 16x4 frag: VGPR0=K=ka, VGPR1=K=ka+1 (contiguous -> vector load)
        float2 av = *(const float2*)(arow + ka);
        v2f a; a.x = av.x; a.y = av.y;
        // B 4x16 frag: VGPR0=row ka, VGPR1=row ka+1, col = n0+lo
        v2f b;
        b.x = B[(size_t)ka * 32 + n0 + lo];
        b.y = B[(size_t)(ka + 1) * 32 + n0 + lo];
        acc = __builtin_amdgcn_wmma_f32_16x16x4_f32(
            /*neg_a=*/false, a, /*neg_b=*/false, b,
            /*c_mod=*/(short)0, acc, /*reuse_a=*/false, /*reuse_b=*/false);
    }

    // C/D layout: VGPR r holds row (r + 8*hi), col = n0+lo
#pragma unroll
    for (int r = 0; r < 8; ++r)
        C[(size_t)(m0 + r + 8 * hi) * 32 + n0 + lo] = acc[r];
}

// ---------------------------------------------------------------------------
// out[i][f] = bias[f] + scale * xw[i][f] * dis[i]^2   (bias + self-loop message)
__global__ void k_self_bias(const float* __restrict__ xw, const float* __restrict__ dis,
                            const float* __restrict__ bias, float* __restrict__ out,
                            float scale, int n) {
    long long idx = (long long)blockIdx.x * blockDim.x + threadIdx.x;
    if (idx >= (long long)n * 32) return;
    int i = (int)(idx >> 5);
    int f = (int)(idx & 31);
    float ds = dis[i];
    out[idx] = bias[f] + scale * xw[idx] * ds * ds;
}

// out[dst][f] += scale * dis[src]*dis[dst] * xw[src][f], one thread per (edge, feature)
__global__ void k_scatter(const int* __restrict__ src, const int* __restrict__ dst,
                          const float* __restrict__ dis, const float* __restrict__ xw,
                          float* __restrict__ out, float scale, long long E) {
    long long idx = (long long)blockIdx.x * blockDim.x + threadIdx.x;
    if (idx >= E * 32) return;
    long long e = idx >> 5;
    int f = (int)(idx & 31);
    int s = src[e];
    int d = dst[e];
    float nrm = scale * dis[s] * dis[d];
    atomicAdd(&out[(long long)d * 32 + f], xw[(long long)s * 32 + f] * nrm);
}

// ELU in place
__global__ void k_elu(float* __restrict__ h, long long n) {
    long long idx = (long long)blockIdx.x * blockDim.x + threadIdx.x;
    if (idx < n) {
        float v = h[idx];
        h[idx] = v > 0.0f ? v : expm1f(v);
    }
}

// ---------------------------------------------------------------------------
extern "C" void kernel_launch(void* const* d_in, const int* in_sizes, int n_in,
                              void* d_out, int out_size, void* d_ws, size_t ws_size,
                              hipStream_t stream) {
    const float* x   = (const float*)d_in[0];   // [N,128]
    const int*   ei  = (const int*)d_in[1];     // [2,E]
    const float* W1  = (const float*)d_in[2];   // [128,32]
    const float* b1  = (const float*)d_in[3];   // [32]
    const float* W2  = (const float*)d_in[4];   // [32,32]
    const float* b2  = (const float*)d_in[5];   // [32]
    float*       out = (float*)d_out;           // [N,32]

    const int N = in_sizes[0] / 128;            // 100000
    const long long E = (long long)(in_sizes[1] / 2); // 1600000
    const int* src = ei;
    const int* dst = ei + E;

    // workspace layout (256B aligned slices)
    auto align256 = [](size_t v) { return (v + 255) & ~(size_t)255; };
    char* w = (char*)d_ws;
    float* dis  = (float*)w;                               // N floats (deg -> dis)
    size_t off  = align256((size_t)N * sizeof(float));
    float* buf1 = (float*)(w + off);                       // N*32: xW1, later hW2
    off += align256((size_t)N * 32 * sizeof(float));
    float* buf2 = (float*)(w + off);                       // N*32: h accumulator / ELU
    (void)ws_size; (void)n_in; (void)out_size;

    const long long NF = (long long)N * 32;
    const long long EF = E * 32;
    dim3 gemm_grid((unsigned)(N / 16), 2);

    // degrees + symmetric normalization
    k_fill_deg<<<(N + TPB - 1) / TPB, TPB, 0, stream>>>(dis, N);
    k_count_deg<<<(unsigned)((E + TPB - 1) / TPB), TPB, 0, stream>>>(dst, dis, (int)E);
    k_rsqrt<<<(N + TPB - 1) / TPB, TPB, 0, stream>>>(dis, N);

    // layer 1: h = elu( propagate(x @ W1) + b1 )
    k_gemm_wmma_f32<<<gemm_grid, 32, 0, stream>>>(x, W1, buf1, 128);
    k_self_bias<<<(unsigned)((NF + TPB - 1) / TPB), TPB, 0, stream>>>(buf1, dis, b1, buf2, 1.0f, N);
    k_scatter<<<(unsigned)((EF + TPB - 1) / TPB), TPB, 0, stream>>>(src, dst, dis, buf1, buf2, 1.0f, E);
    k_elu<<<(unsigned)((NF + TPB - 1) / TPB), TPB, 0, stream>>>(buf2, NF);

    // layer 2: out = EXP_SCALE * propagate(h @ W2) + b2
    k_gemm_wmma_f32<<<gemm_grid, 32, 0, stream>>>(buf2, W2, buf1, 32);
    k_self_bias<<<(unsigned)((NF + TPB - 1) / TPB), TPB, 0, stream>>>(buf1, dis, b2, out, EXP_SCALE, N);
    k_scatter<<<(unsigned)((EF + TPB - 1) / TPB), TPB, 0, stream>>>(src, dst, dis, buf1, out, EXP_SCALE, E);
}